// radar_DynamicEmbedder_90950227460834
// MI455X (gfx1250) — compile-verified
//
#include <hip/hip_runtime.h>
#include <hip/hip_bf16.h>

typedef float v2f __attribute__((ext_vector_type(2)));
typedef float v8f __attribute__((ext_vector_type(8)));

#define HGRID 512
#define WGRID 512
#define CCH   32

// ---------------------------------------------------------------------------
// fp32 WMMA 16x16x4 wrapper (D = A(16x4) * B(4x16) + C, fp32 throughout)
// ---------------------------------------------------------------------------
__device__ __forceinline__ v8f wmma4(v2f a, v2f b, v8f c) {
    return __builtin_amdgcn_wmma_f32_16x16x4_f32(
        /*neg_a=*/false, a, /*neg_b=*/false, b,
        /*c_mod=*/(short)0, c, /*reuse_a=*/false, /*reuse_b=*/false);
}

// ---------------------------------------------------------------------------
// Kernel 1: zero pseudoimage (d_out is poisoned), init hash grid to -1
// ---------------------------------------------------------------------------
__global__ void init_kernel(float4* __restrict__ pseudo, int4* __restrict__ grid) {
    int i = blockIdx.x * blockDim.x + threadIdx.x;
    if (i < (CCH * HGRID * WGRID) / 4) {
        pseudo[i] = make_float4(0.f, 0.f, 0.f, 0.f);
    }
    if (i < (HGRID * WGRID) / 4) {
        grid[i] = make_int4(-1, -1, -1, -1);
    }
}

// ---------------------------------------------------------------------------
// Kernel 2: scatter voxel index into grid (duplicate coords: last-writer-wins,
// matching the undefined-order semantics of jax .at[].set with duplicates)
// ---------------------------------------------------------------------------
__global__ void build_grid_kernel(const int* __restrict__ coors,
                                  int* __restrict__ grid, int N) {
    int n = blockIdx.x * blockDim.x + threadIdx.x;
    if (n < N) {
        int r = coors[2 * n], c = coors[2 * n + 1];
        grid[r * WGRID + c] = n;
    }
}

// ---------------------------------------------------------------------------
// Kernel 3: fused Q/K/V projection, fp32 WMMA.
// One wave per 16-row tile. Six accumulators: {Q,K,V} x {cols 0-15, 16-31}.
// A layout  (16x4 f32): lanes 0-15 -> M=lane, K={k0,k0+1}; lanes 16-31 -> K={k0+2,k0+3}
// B layout  (4x16 f32): lanes 0-15 -> N=lane, K={k0,k0+1}; lanes 16-31 -> K={k0+2,k0+3}
// D layout  (16x16 f32): vgpr r, lanes 0-15 -> (M=r,   N=lane),
//                                lanes 16-31 -> (M=r+8, N=lane-16)
// ---------------------------------------------------------------------------
__global__ __launch_bounds__(256) void
qkv_proj_kernel(const float* __restrict__ X,      // (N,32) voxel_feats
                const float* __restrict__ Wf,     // (96,32) in_proj_w
                const float* __restrict__ Bf,     // (96,)   in_proj_b
                float* __restrict__ qb, float* __restrict__ kb,
                float* __restrict__ vb, int N, int ntiles) {
    int wave = blockIdx.x * (blockDim.x >> 5) + (threadIdx.x >> 5);
    if (wave >= ntiles) return;                 // wave-uniform: EXEC stays all-1s
    int lane = threadIdx.x & 31;
    int nn   = lane & 15;                       // M for A, N for B/D
    int hi   = lane >> 4;

    int row  = wave * 16 + nn;
    int rowc = row < N ? row : (N - 1);
    const float* xp = X + (size_t)rowc * CCH + 2 * hi;

    // Weight row pointers: B[k][col] = W[col][k], col = t*16 + nn
    const float* wp[3][2];
#pragma unroll
    for (int p = 0; p < 3; ++p)
#pragma unroll
        for (int t = 0; t < 2; ++t)
            wp[p][t] = Wf + ((size_t)(p * 32 + t * 16 + nn)) * CCH + 2 * hi;

    v8f acc[3][2];
#pragma unroll
    for (int p = 0; p < 3; ++p)
#pragma unroll
        for (int t = 0; t < 2; ++t)
            acc[p][t] = (v8f){0.f, 0.f, 0.f, 0.f, 0.f, 0.f, 0.f, 0.f};

#pragma unroll
    for (int s = 0; s < 8; ++s) {               // K = 32 in steps of 4
        int k0 = 4 * s;
        v2f a = *(const v2f*)(xp + k0);
#pragma unroll
        for (int p = 0; p < 3; ++p)
#pragma unroll
            for (int t = 0; t < 2; ++t) {
                v2f b = *(const v2f*)(wp[p][t] + k0);
                acc[p][t] = wmma4(a, b, acc[p][t]);
            }
    }

    // Biases for this lane's output columns
    float bias[3][2];
#pragma unroll
    for (int p = 0; p < 3; ++p)
#pragma unroll
        for (int t = 0; t < 2; ++t)
            bias[p][t] = Bf[p * 32 + t * 16 + nn];

    float* outp[3] = {qb, kb, vb};
#pragma unroll
    for (int r = 0; r < 8; ++r) {
        int orow = wave * 16 + r + 8 * hi;
        if (orow < N) {
#pragma unroll
            for (int p = 0; p < 3; ++p) {
                outp[p][(size_t)orow * CCH + nn]      = acc[p][0][r] + bias[p][0];
                outp[p][(size_t)orow * CCH + 16 + nn] = acc[p][1][r] + bias[p][1];
            }
        }
    }
}

// ---------------------------------------------------------------------------
// Kernel 4: attention + output projection + residual + scatter.
// One wave32 per voxel; lane = channel; head = lane>>4 (head_dim = 16).
// ---------------------------------------------------------------------------
__global__ __launch_bounds__(256) void
attn_kernel(const float* __restrict__ X,
            const float* __restrict__ qbuf,
            const float* __restrict__ kbuf,
            const float* __restrict__ vbuf,
            const int* __restrict__ grid,
            const int* __restrict__ coors,
            const float* __restrict__ Wo,   // (32,32) out_proj_w
            const float* __restrict__ bo,   // (32,)
            const float* __restrict__ Bf,   // (96,) in_proj_b (for bk, bv)
            float* __restrict__ pseudo,     // (32,512,512)
            float* __restrict__ out_bev,    // (N,32)
            int N) {
    __shared__ float WoT[CCH * CCH];   // transposed: WoT[c*32 + c'] = Wo[c'][c]
    __shared__ float s_bo[CCH], s_bk[CCH], s_bv[CCH];

    int t = threadIdx.x;
#pragma unroll
    for (int u = 0; u < 4; ++u) {
        int i = t * 4 + u;                         // 256*4 = 1024 elements
        WoT[(i & 31) * CCH + (i >> 5)] = Wo[i];
    }
    if (t < 32)       s_bo[t]      = bo[t];
    else if (t < 64)  s_bk[t - 32] = Bf[32 + (t - 32)];
    else if (t < 96)  s_bv[t - 64] = Bf[64 + (t - 64)];
    __syncthreads();

    int n = blockIdx.x * 8 + (t >> 5);
    if (n >= N) return;
    int lane = t & 31;

    int r0 = coors[2 * n], c0 = coors[2 * n + 1];
    float qc = qbuf[(size_t)n * CCH + lane];

    const int dy[9] = {0, -1, 1, 0, -1, 1, 0, -1, 1};
    const int dx[9] = {0, 0, 0, 1, 1, 1, -1, -1, -1};

    float sc[9], vv[9];
#pragma unroll
    for (int j = 0; j < 9; ++j) {
        int rr = r0 + dy[j], cc = c0 + dx[j];
        int idx = -1;
        if (rr >= 0 && rr < HGRID && cc >= 0 && cc < WGRID)
            idx = grid[rr * WGRID + cc];
        float kc, vc;
        if (idx >= 0) {
            kc = kbuf[(size_t)idx * CCH + lane];
            vc = vbuf[(size_t)idx * CCH + lane];
        } else {                      // zero-feature neighbor -> projection = bias
            kc = s_bk[lane];
            vc = s_bv[lane];
        }
        float p = qc * kc;            // per-head dot: reduce within 16-lane half
        p += __shfl_xor(p, 1, 32);
        p += __shfl_xor(p, 2, 32);
        p += __shfl_xor(p, 4, 32);
        p += __shfl_xor(p, 8, 32);
        sc[j] = p * 0.25f;            // 1/sqrt(16)
        vv[j] = vc;
    }

    float m = sc[0];
#pragma unroll
    for (int j = 1; j < 9; ++j) m = fmaxf(m, sc[j]);
    float l = 0.f, accv = 0.f;
#pragma unroll
    for (int j = 0; j < 9; ++j) {
        float e = __expf(sc[j] - m);
        l += e;
        accv += e * vv[j];
    }
    float outc = accv / l;

    // out projection: att[lane] = bo[lane] + sum_c out[c] * Wo[lane][c]
    float att = s_bo[lane];
#pragma unroll
    for (int c = 0; c < 32; ++c) {
        float oc = __shfl(outc, c, 32);
        att = fmaf(oc, WoT[c * CCH + lane], att);
    }

    float ob = X[(size_t)n * CCH + lane] + att;
    out_bev[(size_t)n * CCH + lane] = ob;
    pseudo[(size_t)lane * (HGRID * WGRID) + r0 * WGRID + c0] = ob;
}

// ---------------------------------------------------------------------------
extern "C" void kernel_launch(void* const* d_in, const int* in_sizes, int n_in,
                              void* d_out, int out_size, void* d_ws, size_t ws_size,
                              hipStream_t stream) {
    const float* X     = (const float*)d_in[0];   // voxel_feats (N,32)
    const float* Wf    = (const float*)d_in[1];   // in_proj_w   (96,32)
    const float* Bf    = (const float*)d_in[2];   // in_proj_b   (96,)
    const float* Wo    = (const float*)d_in[3];   // out_proj_w  (32,32)
    const float* bo    = (const float*)d_in[4];   // out_proj_b  (32,)
    const int*   coors = (const int*)d_in[5];     // voxel_coors (N,2)
    int N = in_sizes[0] / CCH;

    // workspace layout: [grid 512*512 int] [q N*32 f32] [k N*32 f32] [v N*32 f32]
    int*   grid = (int*)d_ws;
    float* qb   = (float*)((char*)d_ws + (size_t)HGRID * WGRID * sizeof(int));
    float* kb   = qb + (size_t)N * CCH;
    float* vb   = kb + (size_t)N * CCH;

    float* pseudo = (float*)d_out;                          // 32*512*512
    float* outbev = pseudo + (size_t)CCH * HGRID * WGRID;   // N*32

    int initN = (CCH * HGRID * WGRID) / 4;  // float4 granularity
    init_kernel<<<(initN + 255) / 256, 256, 0, stream>>>((float4*)pseudo,
                                                         (int4*)grid);
    build_grid_kernel<<<(N + 255) / 256, 256, 0, stream>>>(coors, grid, N);

    int ntiles = (N + 15) / 16;
    int nblk   = (ntiles + 7) / 8;          // 8 waves / block
    qkv_proj_kernel<<<nblk, 256, 0, stream>>>(X, Wf, Bf, qb, kb, vb, N, ntiles);

    attn_kernel<<<(N + 7) / 8, 256, 0, stream>>>(X, qb, kb, vb, grid, coors,
                                                 Wo, bo, Bf, pseudo, outbev, N);
}